// BahdanauAttention_77945066488374
// MI455X (gfx1250) — compile-verified
//
#include <hip/hip_runtime.h>
#include <math.h>

typedef float v2f __attribute__((ext_vector_type(2)));
typedef float v8f __attribute__((ext_vector_type(8)));

#define Bsz 8
#define Tsz 512
#define Ssz 512
#define Hsz 128

#if __has_builtin(__builtin_amdgcn_global_load_async_to_lds_b128) && \
    __has_builtin(__builtin_amdgcn_global_load_async_to_lds_b32)  && \
    __has_builtin(__builtin_amdgcn_s_wait_asynccnt)
#define USE_ASYNC_LDS 1
typedef int v4i __attribute__((vector_size(16)));
typedef __attribute__((address_space(1))) v4i* g_v4i_t;   // global int4*
typedef __attribute__((address_space(3))) v4i* l_v4i_t;   // LDS    int4*
typedef __attribute__((address_space(1))) int* g_i32_t;   // global int*
typedef __attribute__((address_space(3))) int* l_i32_t;   // LDS    int*
#endif

// ---------------------------------------------------------------------------
// Kernel 1: Y(rows x 128) = X(rows x 128) @ W^T(128 x 128), fp32 WMMA 16x16x4.
// grid.x = rows/16 (M tiles); block = 256 (8 waves), wave w owns N-block w*16.
// ---------------------------------------------------------------------------
__global__ __launch_bounds__(256) void proj_wmma_kernel(
    const float* __restrict__ X, const float* __restrict__ W,
    float* __restrict__ Y) {
  const int m0   = blockIdx.x * 16;
  const int wave = threadIdx.x >> 5;
  const int lane = threadIdx.x & 31;
  const int n0   = wave * 16;
  const int M    = lane & 15;            // A row / B col for this lane
  const int bk   = (lane >> 4) << 1;     // lanes 0-15: K=0,1 ; lanes 16-31: K=2,3

  const float* __restrict__ xrow = X + (m0 + M) * Hsz;
  const float* __restrict__ wrow = W + (n0 + M) * Hsz;   // B[k][n] = W[n][k]

  v8f c = {};
  #pragma unroll 8
  for (int k0 = 0; k0 < Hsz; k0 += 4) {
    v2f a, b;
    a.x = xrow[k0 + bk];  a.y = xrow[k0 + bk + 1];
    b.x = wrow[k0 + bk];  b.y = wrow[k0 + bk + 1];
    c = __builtin_amdgcn_wmma_f32_16x16x4_f32(false, a, false, b,
                                              (short)0, c, false, false);
  }
  // C/D layout: VGPR j -> M = j (lanes 0-15) or j+8 (lanes 16-31), N = lane&15
  const int n  = n0 + M;
  const int mb = m0 + ((lane >> 4) << 3);
  #pragma unroll
  for (int j = 0; j < 8; ++j) Y[(mb + j) * Hsz + n] = c[j];
}

// ---------------------------------------------------------------------------
// Kernel 2: fused scores + mask-quirk + softmax + ct GEMM.
// grid.x = B * T/16 ; block = 512 (16 waves). Thread s=tid owns one key.
// ---------------------------------------------------------------------------
__global__ __launch_bounds__(512) void attn_ct_kernel(
    const float* __restrict__ gws,   // (B,T,H)  query @ W_s^T
    const float* __restrict__ gwh,   // (B,S,H)  enc   @ W_h^T
    const float* __restrict__ enc,   // (B,S,H)
    const float* __restrict__ v,     // (H)
    const int*   __restrict__ slen,  // (B)
    float* __restrict__ ctb) {       // (B,T,H) out
  __shared__ float ws_l[16 * Hsz];   //  8 KB
  __shared__ float v_l[Hsz];         // .5 KB
  __shared__ float sc[16 * Ssz];     // 32 KB  scores -> attn tile

  const int b   = blockIdx.x >> 5;          // 32 T-tiles per batch
  const int t0  = (blockIdx.x & 31) * 16;
  const int tid = threadIdx.x;

  const float* __restrict__ wsg = gws + (b * Tsz + t0) * Hsz;
#ifdef USE_ASYNC_LDS
  // gfx1250 async copy: global -> LDS without passing through VGPRs.
  // 512 threads x b128 = exactly the 16x128 fp32 tile.
  {
    float* wsg_nc = const_cast<float*>(wsg);
    __builtin_amdgcn_global_load_async_to_lds_b128(
        (g_v4i_t)(wsg_nc + tid * 4), (l_v4i_t)(ws_l + tid * 4), 0, 0);
    if (tid < Hsz) {   // waves 0-3 uniformly: stage v
      float* v_nc = const_cast<float*>(v);
      __builtin_amdgcn_global_load_async_to_lds_b32(
          (g_i32_t)(v_nc + tid), (l_i32_t)(v_l + tid), 0, 0);
    }
    __builtin_amdgcn_s_wait_asynccnt(0);
    __syncthreads();
  }
#else
  for (int i = tid; i < 16 * Hsz; i += 512) ws_l[i] = wsg[i];
  if (tid < Hsz) v_l[tid] = v[tid];
  __syncthreads();
#endif

  // ---- additive scores: acc[t] = sum_h v[h] * tanh(ws[t][h] + wh[s][h]) ----
  const int s = tid;
  float acc[16];
  #pragma unroll
  for (int t = 0; t < 16; ++t) acc[t] = 0.0f;

  const float4* __restrict__ whrow =
      reinterpret_cast<const float4*>(gwh + (b * Ssz + s) * Hsz);
  const float4* __restrict__ ws4 = reinterpret_cast<const float4*>(ws_l);
  const float4* __restrict__ v4p = reinterpret_cast<const float4*>(v_l);

  for (int hh = 0; hh < Hsz / 4; ++hh) {
    if (hh + 4 < Hsz / 4)                       // global_prefetch_b8 ahead
      __builtin_prefetch((const void*)&whrow[hh + 4], 0, 0);
    const float4 w4 = whrow[hh];
    const float4 vv = v4p[hh];                  // one ds_load_b128, broadcast
    #pragma unroll
    for (int t = 0; t < 16; ++t) {
      const float4 wst = ws4[t * (Hsz / 4) + hh];  // ds_load_b128, broadcast
      acc[t] += vv.x * tanhf(wst.x + w4.x);
      acc[t] += vv.y * tanhf(wst.y + w4.y);
      acc[t] += vv.z * tanhf(wst.z + w4.z);
      acc[t] += vv.w * tanhf(wst.w + w4.w);
    }
  }

  // faithful quirk: zero (not -inf) masked scores, keep them inside softmax
  const bool keep = s < slen[b];
  #pragma unroll
  for (int t = 0; t < 16; ++t) sc[t * Ssz + s] = keep ? acc[t] : 0.0f;
  __syncthreads();

  // ---- softmax over S: wave w owns row t=w, wave32 shuffle reductions ----
  const int wave = tid >> 5;
  const int lane = tid & 31;
  {
    float* __restrict__ row = sc + wave * Ssz;
    float vals[16];
    float mx = -3.402823466e38f;
    #pragma unroll
    for (int i = 0; i < 16; ++i) {
      vals[i] = row[lane + 32 * i];
      mx = fmaxf(mx, vals[i]);
    }
    #pragma unroll
    for (int off = 16; off > 0; off >>= 1)
      mx = fmaxf(mx, __shfl_xor(mx, off, 32));
    float sum = 0.0f;
    #pragma unroll
    for (int i = 0; i < 16; ++i) { vals[i] = expf(vals[i] - mx); sum += vals[i]; }
    #pragma unroll
    for (int off = 16; off > 0; off >>= 1)
      sum += __shfl_xor(sum, off, 32);
    const float inv = 1.0f / sum;
    #pragma unroll
    for (int i = 0; i < 16; ++i) row[lane + 32 * i] = vals[i] * inv;
  }
  __syncthreads();

  // ---- ct tile (16 x 128) = attn(16 x 512, LDS) @ enc(512 x 128), WMMA ----
  const int wv_u = __builtin_amdgcn_readfirstlane(wave);  // scalar branch,
  if (wv_u < 8) {                                         // EXEC stays all-1s
    const int h0 = wv_u * 16;
    const int M  = lane & 15;
    const int bk = (lane >> 4) << 1;
    const float* __restrict__ encb = enc + b * Ssz * Hsz;
    v8f c = {};
    for (int k0 = 0; k0 < Ssz; k0 += 4) {
      const int k = k0 + bk;
      v2f a, bb;
      a.x  = sc[M * Ssz + k];                // A from LDS (ds_load)
      a.y  = sc[M * Ssz + k + 1];
      bb.x = encb[k * Hsz + h0 + M];         // B from global
      bb.y = encb[(k + 1) * Hsz + h0 + M];
      c = __builtin_amdgcn_wmma_f32_16x16x4_f32(false, a, false, bb,
                                               (short)0, c, false, false);
    }
    const int n  = h0 + M;
    const int mb = t0 + ((lane >> 4) << 3);
    #pragma unroll
    for (int j = 0; j < 8; ++j)
      ctb[(b * Tsz + mb + j) * Hsz + n] = c[j];
  }
}

// ---------------------------------------------------------------------------
// Kernel 3: out = tanh(concat[ct,q](4096 x 256) @ W_out^T(256->128) + bias)
// ---------------------------------------------------------------------------
__global__ __launch_bounds__(256) void out_wmma_kernel(
    const float* __restrict__ ct, const float* __restrict__ q,
    const float* __restrict__ Wo, const float* __restrict__ bo,
    float* __restrict__ out) {
  const int m0   = blockIdx.x * 16;
  const int wave = threadIdx.x >> 5;
  const int lane = threadIdx.x & 31;
  const int n0   = wave * 16;
  const int M    = lane & 15;
  const int bk   = (lane >> 4) << 1;

  const float* __restrict__ act  = ct + (m0 + M) * Hsz;
  const float* __restrict__ aq   = q  + (m0 + M) * Hsz - Hsz;  // offset for k>=H
  const float* __restrict__ brow = Wo + (n0 + M) * (2 * Hsz);

  v8f c = {};
  for (int k0 = 0; k0 < 2 * Hsz; k0 += 4) {
    const int k = k0 + bk;                    // k,k+1 never straddle H (even k)
    const float* __restrict__ asrc = (k < Hsz) ? act : aq;
    v2f a, b;
    a.x = asrc[k];   a.y = asrc[k + 1];
    b.x = brow[k];   b.y = brow[k + 1];
    c = __builtin_amdgcn_wmma_f32_16x16x4_f32(false, a, false, b,
                                              (short)0, c, false, false);
  }
  const int   n    = n0 + M;
  const int   mb   = m0 + ((lane >> 4) << 3);
  const float bias = bo[n];
  #pragma unroll
  for (int j = 0; j < 8; ++j)
    out[(mb + j) * Hsz + n] = tanhf(c[j] + bias);
}

// ---------------------------------------------------------------------------
extern "C" void kernel_launch(void* const* d_in, const int* in_sizes, int n_in,
                              void* d_out, int out_size, void* d_ws, size_t ws_size,
                              hipStream_t stream) {
  const float* query = (const float*)d_in[0];   // (B,T,H)
  const float* enc   = (const float*)d_in[1];   // (B,S,H)
  const int*   slen  = (const int*)  d_in[2];   // (B)
  const float* W_h   = (const float*)d_in[3];   // (H,H)
  const float* W_s   = (const float*)d_in[4];   // (H,H)
  const float* v     = (const float*)d_in[5];   // (H)
  const float* W_ow  = (const float*)d_in[6];   // (H,2H)
  const float* W_ob  = (const float*)d_in[7];   // (H)
  float* out = (float*)d_out;

  const size_t N = (size_t)Bsz * Tsz * Hsz;     // 524288 floats
  float* gws = (float*)d_ws;                    // WS  (B,T,H)
  float* gwh = gws + N;                         // WH  (B,S,H)
  float* ctb = gwh + N;                         // ct  (B,T,H)

  proj_wmma_kernel<<<(Bsz * Tsz) / 16, 256, 0, stream>>>(query, W_s, gws);
  proj_wmma_kernel<<<(Bsz * Ssz) / 16, 256, 0, stream>>>(enc,   W_h, gwh);
  attn_ct_kernel  <<<(Bsz * Tsz) / 16, 512, 0, stream>>>(gws, gwh, enc, v, slen, ctb);
  out_wmma_kernel <<<(Bsz * Tsz) / 16, 256, 0, stream>>>(ctb, query, W_ow, W_ob, out);
}